// TaxaNetLoss_48249662604031
// MI455X (gfx1250) — compile-verified
//
#include <hip/hip_runtime.h>
#include <math.h>

// Taxonomy constants (fixed by the reference).
#define C_TOTAL 6175
#define ROW_BYTES (C_TOTAL * 4)          // 24700 B
#define ROW_B128_BYTES 24576             // 6 * 256 lanes * 16 B
#define NTHREADS 256

__global__ __launch_bounds__(NTHREADS)
void taxa_row_kernel(const float* __restrict__ y_pred,
                     const int*   __restrict__ y_true,
                     const float* __restrict__ H,
                     const int*   __restrict__ parent,
                     float*       __restrict__ ws,   // [7 * B] : ce0..ce3, pen1..pen3
                     int B)
{
    __shared__ float s_row[C_TOTAL + 1];
    __shared__ float s_rv[8];
    __shared__ int   s_ra[8];
    __shared__ float s_bmax;
    __shared__ int   s_argm[4];
    __shared__ int   s_lab[4];

    const int tid  = threadIdx.x;
    const int row  = blockIdx.x;
    const int wid  = tid >> 5;
    const int lane = tid & 31;

    // ---------------- async stage: global row -> LDS (CDNA5 async-to-LDS path) ----
    {
        unsigned long long gbase = (unsigned long long)(y_pred + (long long)row * C_TOTAL);
        // AS(3) offset: low 32 bits of the generic shared pointer are the LDS byte offset.
        unsigned lds0 = (unsigned)(unsigned long long)&s_row[0];

        // 6 x b128 per thread: covers bytes [0, 24576)
        #pragma unroll
        for (int c = 0; c < 6; ++c) {
            unsigned off  = (unsigned)(c * (NTHREADS * 16) + tid * 16);
            unsigned ldsa = lds0 + off;
            unsigned long long ga = gbase + off;
            asm volatile("global_load_async_to_lds_b128 %0, %1, off"
                         :: "v"(ldsa), "v"(ga) : "memory");
        }
        // tail: 31 dwords, bytes [24576, 24700)
        if (tid < 31) {
            unsigned off  = ROW_B128_BYTES + (unsigned)tid * 4u;
            unsigned ldsa = lds0 + off;
            unsigned long long ga = gbase + off;
            asm volatile("global_load_async_to_lds_b32 %0, %1, off"
                         :: "v"(ldsa), "v"(ga) : "memory");
        }
        asm volatile("s_wait_asynccnt 0" ::: "memory");
    }

    // Ancestor chain (labels are per-level local indices).
    if (tid == 0) {
        int g3 = y_true[row] + 1375;
        int g2 = parent[g3];
        int g1 = parent[g2];
        int g0 = parent[g1];
        s_lab[0] = g0;
        s_lab[1] = g1 - 25;
        s_lab[2] = g2 - 175;
        s_lab[3] = g3 - 1375;
    }
    __syncthreads();   // LDS row visible to all waves; labels published

    const int LO[4] = {0, 25, 175, 1375};
    const int NN[4] = {25, 150, 1200, 4800};

    #pragma unroll 1
    for (int k = 0; k < 4; ++k) {
        const int lo = LO[k];
        const int n  = NN[k];

        // ---- max + argmax (first-occurrence tie-break, like jnp.argmax) ----
        float lv = -__builtin_inff();
        int   la = 0x7fffffff;
        for (int i = tid; i < n; i += NTHREADS) {
            float v = s_row[lo + i];
            if (v > lv) { lv = v; la = i; }   // strict > keeps lowest index per thread
        }
        #pragma unroll
        for (int off = 16; off > 0; off >>= 1) {
            float ov = __shfl_down(lv, off, 32);
            int   oa = __shfl_down(la, off, 32);
            if (ov > lv || (ov == lv && oa < la)) { lv = ov; la = oa; }
        }
        if (lane == 0) { s_rv[wid] = lv; s_ra[wid] = la; }
        __syncthreads();
        if (tid == 0) {
            float bv = s_rv[0]; int ba = s_ra[0];
            #pragma unroll
            for (int w = 1; w < 8; ++w) {
                if (s_rv[w] > bv || (s_rv[w] == bv && s_ra[w] < ba)) {
                    bv = s_rv[w]; ba = s_ra[w];
                }
            }
            s_bmax    = bv;
            s_argm[k] = ba;
        }
        __syncthreads();
        const float M = s_bmax;

        // ---- sum of exp(x - M) ----
        float ls = 0.0f;
        for (int i = tid; i < n; i += NTHREADS)
            ls += expf(s_row[lo + i] - M);
        #pragma unroll
        for (int off = 16; off > 0; off >>= 1)
            ls += __shfl_down(ls, off, 32);
        if (lane == 0) s_rv[wid] = ls;
        __syncthreads();
        if (tid == 0) {
            float tot = 0.0f;
            #pragma unroll
            for (int w = 0; w < 8; ++w) tot += s_rv[w];
            int   lab = s_lab[k];
            float ce  = -(s_row[lo + lab] - M - logf(tot));
            ws[(size_t)k * B + row] = ce;
        }
        __syncthreads();   // protect s_rv / s_bmax reuse next level
    }

    // ---- hierarchy penalties: gather H[argm[k-1], argm[k]] (local-index quirk kept) ----
    if (tid == 0) {
        const float Ec = 2.718281828459045f;   // float(np.e)
        int a0 = s_argm[0], a1 = s_argm[1], a2 = s_argm[2], a3 = s_argm[3];
        float h1 = H[(size_t)a0 * C_TOTAL + a1];
        float h2 = H[(size_t)a1 * C_TOTAL + a2];
        float h3 = H[(size_t)a2 * C_TOTAL + a3];
        ws[(size_t)4 * B + row] = (h1 == 0.0f) ? Ec : 0.0f;
        ws[(size_t)5 * B + row] = (h2 == 0.0f) ? Ec : 0.0f;
        ws[(size_t)6 * B + row] = (h3 == 0.0f) ? Ec : 0.0f;
    }
}

// Deterministic fixed-order reduction of the 7 per-row arrays + scalar combine.
__global__ __launch_bounds__(NTHREADS)
void taxa_finalize_kernel(const float* __restrict__ ws, float* __restrict__ out, int B)
{
    __shared__ float sp[NTHREADS];
    __shared__ float s_res[7];
    const int tid = threadIdx.x;

    for (int q = 0; q < 7; ++q) {
        float acc = 0.0f;
        for (int i = tid; i < B; i += NTHREADS)
            acc += ws[(size_t)q * B + i];
        sp[tid] = acc;
        __syncthreads();
        #pragma unroll
        for (int s = NTHREADS / 2; s > 0; s >>= 1) {
            if (tid < s) sp[tid] += sp[tid + s];
            __syncthreads();
        }
        if (tid == 0) s_res[q] = sp[0];
        __syncthreads();
    }

    if (tid == 0) {
        const float invB = 1.0f / (float)B;
        const float W[4] = {0.25f, 0.25f, 0.15f, 0.10f};
        float s    = s_res[0] * invB;   // CE level 0 (mean)
        float loss = 0.0f;
        #pragma unroll
        for (int k = 1; k < 4; ++k) {
            // s = (s + pen_k + CE_k) * W[k]; loss += s
            s = (s + s_res[3 + k] + s_res[k] * invB) * W[k];
            loss += s;
        }
        out[0] = loss;
    }
}

extern "C" void kernel_launch(void* const* d_in, const int* in_sizes, int n_in,
                              void* d_out, int out_size, void* d_ws, size_t ws_size,
                              hipStream_t stream) {
    const float* y_pred = (const float*)d_in[0];
    const int*   y_true = (const int*)  d_in[1];
    const float* H      = (const float*)d_in[2];
    const int*   parent = (const int*)  d_in[3];
    float* ws  = (float*)d_ws;
    float* out = (float*)d_out;
    const int B = in_sizes[1];   // batch = #labels

    taxa_row_kernel<<<B, NTHREADS, 0, stream>>>(y_pred, y_true, H, parent, ws, B);
    taxa_finalize_kernel<<<1, NTHREADS, 0, stream>>>(ws, out, B);
}